// Model_G_60842506715229
// MI455X (gfx1250) — compile-verified
//
#include <hip/hip_runtime.h>

typedef __attribute__((ext_vector_type(2))) float v2f;
typedef __attribute__((ext_vector_type(8))) float v8f;

#define SLOPE 0.2f

__device__ __forceinline__ void atomicMaxFloat(float* addr, float val) {
    // works when *addr initialized to -inf (0xFF800000)
    if (val >= 0.0f) atomicMax((int*)addr, __float_as_int(val));
    else             atomicMin((unsigned int*)addr, __float_as_uint(val));
}

// ---------------- init: zero accumulators, seed out with out_b ----------------
__global__ void k_init(float* mmax, float* denom, float* agg, float* smax, float* esum,
                       float* out, const float* __restrict__ out_b, int NH, int aggN) {
    int i = blockIdx.x * blockDim.x + threadIdx.x;
    if (i < aggN) agg[i] = 0.0f;
    if (i < NH) { mmax[i] = -__builtin_huge_valf(); denom[i] = 0.0f; }
    if (i == 0) {
        smax[0] = -__builtin_huge_valf(); esum[0] = 0.0f;
        out[0] = out_b[0]; out[1] = out_b[1]; out[2] = out_b[2]; out[3] = 0.0f;
    }
}

// ---------------- h = x @ fc_w^T + fc_b  (WMMA f32 16x16x4, K=128) ------------
__global__ void k_embed(const float* __restrict__ x, const float* __restrict__ fc_w,
                        const float* __restrict__ fc_b, float* __restrict__ hbuf) {
    int tile = blockIdx.x;                 // N/16 tiles, one wave32 per tile
    int lane = threadIdx.x;
    int r  = lane & 15;
    int kh = (lane >> 4) << 1;             // 0 or 2 (K sub-offset per ISA A/B layout)
    const float* xr = x    + (long)(tile * 16 + r) * 128;  // A: row = M
    const float* wr = fc_w + (long)r * 128;                // B[k,n] = fc_w[n,k]
    v8f acc = {};
    for (int k = 0; k < 128; k += 4) {
        v2f a, b;
        a[0] = xr[k + kh];  a[1] = xr[k + kh + 1];
        b[0] = wr[k + kh];  b[1] = wr[k + kh + 1];
        acc = __builtin_amdgcn_wmma_f32_16x16x4_f32(false, a, false, b, (short)0, acc,
                                                    false, false);
    }
    float bias = fc_b[r];
    int mbase = tile * 16 + ((lane >> 4) << 3);  // lanes>=16 hold M=v+8
#pragma unroll
    for (int v = 0; v < 8; ++v)
        hbuf[(mbase + v) * 16 + r] = acc[v] + bias;
}

// ---------------- xp = h @ lin_w^T  (WMMA f32 16x16x4, K=16) ------------------
__global__ void k_xform(const float* __restrict__ hbuf, const float* __restrict__ lin_w,
                        float* __restrict__ xp) {
    int tile = blockIdx.x;                 // row tile
    int ht   = blockIdx.y;                 // head = 16-col tile
    int lane = threadIdx.x;
    int r  = lane & 15;
    int kh = (lane >> 4) << 1;
    const float* hr = hbuf  + (long)(tile * 16 + r) * 16;
    const float* wr = lin_w + (long)(ht * 16 + r) * 16;    // B[k,c] = lin_w[ht*16+c, k]
    v8f acc = {};
#pragma unroll
    for (int k = 0; k < 16; k += 4) {
        v2f a, b;
        a[0] = hr[k + kh];  a[1] = hr[k + kh + 1];
        b[0] = wr[k + kh];  b[1] = wr[k + kh + 1];
        acc = __builtin_amdgcn_wmma_f32_16x16x4_f32(false, a, false, b, (short)0, acc,
                                                    false, false);
    }
    int mbase = tile * 16 + ((lane >> 4) << 3);
#pragma unroll
    for (int v = 0; v < 8; ++v)
        xp[(long)(mbase + v) * 128 + ht * 16 + r] = acc[v];
}

// ---------------- a_src / a_dst : per (node, head) 16-dot -----------------------
__global__ void k_attcoef(const float* __restrict__ xp, const float* __restrict__ att_src,
                          const float* __restrict__ att_dst, float* __restrict__ a_src,
                          float* __restrict__ a_dst, int NH) {
    int i = blockIdx.x * blockDim.x + threadIdx.x;
    if (i >= NH) return;
    int n = i >> 3, hh = i & 7;
    const float* v = xp + (long)n * 128 + hh * 16;
    float s = 0.0f, d = 0.0f;
#pragma unroll
    for (int c = 0; c < 16; ++c) {
        s += v[c] * att_src[hh * 16 + c];
        d += v[c] * att_dst[hh * 16 + c];
    }
    a_src[i] = s; a_dst[i] = d;
}

// ---------------- W2[h,d] = sum_c lin_edge_w[h*16+c,d] * att_edge[h,c] ---------
__global__ void k_w2(const float* __restrict__ lin_edge_w, const float* __restrict__ att_edge,
                     float* __restrict__ W2) {
    int i = threadIdx.x;                   // 64 threads
    int hh = i >> 3, d = i & 7;
    float s = 0.0f;
#pragma unroll
    for (int c = 0; c < 16; ++c)
        s += lin_edge_w[(hh * 16 + c) * 8 + d] * att_edge[hh * 16 + c];
    W2[i] = s;
}

// ---------------- logits + LeakyReLU + segment max -----------------------------
__global__ void k_logits(const float* __restrict__ edge_attr, const int* __restrict__ ei,
                         const float* __restrict__ a_src, const float* __restrict__ a_dst,
                         const float* __restrict__ W2, float* __restrict__ logits,
                         float* mmax, int E) {
    int e = blockIdx.x * blockDim.x + threadIdx.x;
    if (e >= E) return;
    int s = ei[e], d = ei[E + e];
    float ea[8];
#pragma unroll
    for (int j = 0; j < 8; ++j) ea[j] = edge_attr[(long)e * 8 + j];
#pragma unroll
    for (int hh = 0; hh < 8; ++hh) {
        float ae = 0.0f;
#pragma unroll
        for (int j = 0; j < 8; ++j) ae += ea[j] * W2[hh * 8 + j];
        float l = a_src[s * 8 + hh] + a_dst[d * 8 + hh] + ae;
        l = (l >= 0.0f) ? l : SLOPE * l;
        logits[(long)e * 8 + hh] = l;
        atomicMaxFloat(&mmax[d * 8 + hh], l);
    }
}

// ---------------- exp(logit - m[dst]) and segment sum --------------------------
__global__ void k_expden(const int* __restrict__ ei, const float* __restrict__ mmax,
                         float* __restrict__ logits, float* denom, int E) {
    int e = blockIdx.x * blockDim.x + threadIdx.x;
    if (e >= E) return;
    int d = ei[E + e];
#pragma unroll
    for (int hh = 0; hh < 8; ++hh) {
        float ex = __expf(logits[(long)e * 8 + hh] - mmax[d * 8 + hh]);
        logits[(long)e * 8 + hh] = ex;     // reuse buffer: now holds exp values
        atomicAdd(&denom[d * 8 + hh], ex);
    }
}

// ---------------- alpha * xp[src] scatter-add to agg[dst]; probe attn66 --------
__global__ void k_scatter(const int* __restrict__ ei, const float* __restrict__ xp,
                          const float* __restrict__ exbuf, const float* __restrict__ denom,
                          float* agg, float* attn66, int E) {
    int i = blockIdx.x * blockDim.x + threadIdx.x;
    if (i >= E * 8) return;
    int e = i >> 3, hh = i & 7;
    int s = ei[e], d = ei[E + e];
    float alpha = exbuf[(long)e * 8 + hh] / (denom[d * 8 + hh] + 1e-16f);
    if (hh == 0 && s == 6 && d == 6) atomicAdd(attn66, alpha);
    const float* xv = xp + (long)s * 128 + hh * 16;
    float* av = agg + (long)d * 128 + hh * 16;
#pragma unroll
    for (int c = 0; c < 16; ++c) atomicAdd(&av[c], alpha * xv[c]);
}

// ---------------- head mean + bias + ELU ---------------------------------------
__global__ void k_nodeout(const float* __restrict__ agg, const float* __restrict__ conv_bias,
                          float* __restrict__ outf, int N) {
    int i = blockIdx.x * blockDim.x + threadIdx.x;
    if (i >= N * 16) return;
    int n = i >> 4, c = i & 15;
    float s = 0.0f;
#pragma unroll
    for (int hh = 0; hh < 8; ++hh) s += agg[(long)n * 128 + hh * 16 + c];
    float o = s * 0.125f + conv_bias[c];
    o = (o > 0.0f) ? o : (__expf(o) - 1.0f);   // ELU(alpha=1)
    outf[i] = o;
}

// ---------------- per-node score + global max ----------------------------------
__global__ void k_score(const float* __restrict__ outf, const float* __restrict__ afw,
                        const float* __restrict__ afb, float* __restrict__ score,
                        float* smax, int N) {
    int n = blockIdx.x * blockDim.x + threadIdx.x;
    if (n >= N) return;
    float s = afb[0];
#pragma unroll
    for (int c = 0; c < 16; ++c) s += outf[n * 16 + c] * afw[c];
    score[n] = s;
    atomicMaxFloat(smax, s);
}

__global__ void k_esum(const float* __restrict__ score, const float* __restrict__ smax,
                       float* esum, int N) {
    int n = blockIdx.x * blockDim.x + threadIdx.x;
    if (n >= N) return;
    atomicAdd(esum, __expf(score[n] - smax[0]));
}

// ---------------- y[o] = sum_n w[n] * <outf[n,:], out_w[o, n*16:]> -------------
__global__ void k_final(const float* __restrict__ outf, const float* __restrict__ score,
                        const float* __restrict__ smax, const float* __restrict__ esum,
                        const float* __restrict__ out_w, float* out, int N) {
    int n = blockIdx.x * blockDim.x + threadIdx.x;
    if (n >= N) return;
    float w = __expf(score[n] - smax[0]) / esum[0];
#pragma unroll
    for (int o = 0; o < 3; ++o) {
        const float* wr = out_w + (long)o * ((long)N * 16) + (long)n * 16;
        float s = 0.0f;
#pragma unroll
        for (int c = 0; c < 16; ++c) s += outf[n * 16 + c] * wr[c];
        atomicAdd(&out[o], w * s);
    }
}

extern "C" void kernel_launch(void* const* d_in, const int* in_sizes, int n_in,
                              void* d_out, int out_size, void* d_ws, size_t ws_size,
                              hipStream_t stream) {
    const float* x          = (const float*)d_in[0];
    const float* edge_attr  = (const float*)d_in[1];
    const float* fc_w       = (const float*)d_in[2];
    const float* fc_b       = (const float*)d_in[3];
    const float* lin_w      = (const float*)d_in[4];
    const float* att_src    = (const float*)d_in[5];
    const float* att_dst    = (const float*)d_in[6];
    const float* att_edge   = (const float*)d_in[7];
    const float* lin_edge_w = (const float*)d_in[8];
    const float* conv_bias  = (const float*)d_in[9];
    const float* attn_fc_w  = (const float*)d_in[10];
    const float* attn_fc_b  = (const float*)d_in[11];
    const float* out_w      = (const float*)d_in[12];
    const float* out_b      = (const float*)d_in[13];
    const int*   ei         = (const int*)d_in[14];

    const int N = in_sizes[0] / 128;   // 10000 (multiple of 16)
    const int E = in_sizes[14] / 2;    // 640000

    // workspace layout (floats)
    float* W = (float*)d_ws;
    size_t off = 0;
    float* hbuf   = W + off; off += (size_t)N * 16;
    float* xp     = W + off; off += (size_t)N * 128;
    float* a_src  = W + off; off += (size_t)N * 8;
    float* a_dst  = W + off; off += (size_t)N * 8;
    float* W2     = W + off; off += 64;
    float* logits = W + off; off += (size_t)E * 8;
    float* mmax   = W + off; off += (size_t)N * 8;
    float* denom  = W + off; off += (size_t)N * 8;
    float* agg    = W + off; off += (size_t)N * 128;
    float* outf   = W + off; off += (size_t)N * 16;
    float* score  = W + off; off += (size_t)N;
    float* smax   = W + off; off += 1;
    float* esum   = W + off; off += 1;

    float* out = (float*)d_out;   // [y0,y1,y2, attn66]
    const int NH = N * 8, aggN = N * 128;

    k_init   <<<(aggN + 255) / 256, 256, 0, stream>>>(mmax, denom, agg, smax, esum, out, out_b, NH, aggN);
    k_embed  <<<N / 16, 32, 0, stream>>>(x, fc_w, fc_b, hbuf);
    k_xform  <<<dim3(N / 16, 8), 32, 0, stream>>>(hbuf, lin_w, xp);
    k_attcoef<<<(NH + 255) / 256, 256, 0, stream>>>(xp, att_src, att_dst, a_src, a_dst, NH);
    k_w2     <<<1, 64, 0, stream>>>(lin_edge_w, att_edge, W2);
    k_logits <<<(E + 255) / 256, 256, 0, stream>>>(edge_attr, ei, a_src, a_dst, W2, logits, mmax, E);
    k_expden <<<(E + 255) / 256, 256, 0, stream>>>(ei, mmax, logits, denom, E);
    k_scatter<<<((E * 8) + 255) / 256, 256, 0, stream>>>(ei, xp, logits, denom, agg, out + 3, E);
    k_nodeout<<<(N * 16 + 255) / 256, 256, 0, stream>>>(agg, conv_bias, outf, N);
    k_score  <<<(N + 255) / 256, 256, 0, stream>>>(outf, attn_fc_w, attn_fc_b, score, smax, N);
    k_esum   <<<(N + 255) / 256, 256, 0, stream>>>(score, smax, esum, N);
    k_final  <<<(N + 255) / 256, 256, 0, stream>>>(outf, score, smax, esum, out_w, out, N);
}